// SimpleMLP_66005057405514
// MI455X (gfx1250) — compile-verified
//
#include <hip/hip_runtime.h>

// CDNA5 (gfx1250) FFT-MLP: radix-16^3 4096-pt FFT per batch row using
// v_wmma_f32_16x16x4_f32 for the 16-point DFT stages; x-row streamed into
// LDS with global_load_async_to_lds_b128 (ASYNCcnt); twiddles via
// incremental complex rotations (no per-element sincos).

typedef __attribute__((ext_vector_type(2))) float v2f;
typedef __attribute__((ext_vector_type(8))) float v8f;

#define PI2F 6.28318530717958647692f

static __device__ __forceinline__ v8f wmma4(v2f a, v2f b, v8f c) {
  // 8 args: (neg_a, A, neg_b, B, c_mod, C, reuse_a, reuse_b)
  return __builtin_amdgcn_wmma_f32_16x16x4_f32(false, a, false, b, (short)0, c,
                                               false, false);
}

// C += A*B for complex 16x16 tiles, K-chunk of 4. f32 WMMA has no A/B negate
// (ISA: NEG = CNeg only), so negate Ai in VALU for the real part.
static __device__ __forceinline__ void cmac4(v8f& cr, v8f& ci, v2f ar, v2f ai,
                                             v2f br, v2f bi) {
  v2f nai; nai.x = -ai.x; nai.y = -ai.y;
  cr = wmma4(ar, br, cr);
  cr = wmma4(nai, bi, cr);
  ci = wmma4(ar, bi, ci);
  ci = wmma4(ai, br, ci);
}

__global__ __launch_bounds__(256) void fftmlp_kernel(
    const float* __restrict__ x, const float* __restrict__ w1,
    const float* __restrict__ w2, float* __restrict__ out) {
  __shared__ float smem[16384];          // 64 KB: 2x (re,im) planes of 4096
  float* s_re = smem;                    // buffer S
  float* s_im = smem + 4096;
  float* t_re = smem + 8192;             // buffer T (also raw-x landing zone)
  float* t_im = smem + 12288;

  const int tid  = threadIdx.x;
  const int lane = tid & 31;
  const int wid  = tid >> 5;
  const int half = lane >> 4;            // 0: lanes 0-15, 1: lanes 16-31
  const int mi   = lane & 15;
  const long long row = blockIdx.x;

  // --- Kick off async DMA of the raw x row (32 KB) into LDS bytes of the T
  // --- buffer. B128 moves 16 B/lane; 8 waves x 8 iters x 512 B = 32 KB.
  // --- LDS aperture maps addr[31:0] directly (ISA 10.2), so the low 32 bits
  // --- of the generic shared pointer are the LDS byte offset.
  {
    const float* xrow = x + row * 8192;
    unsigned ldsbase = (unsigned)(unsigned long long)(const void*)(smem + 8192);
#pragma unroll
    for (int i = 0; i < 8; ++i) {
      int c = i * 256 + tid;             // 16-byte chunk id, 2048 chunks
      unsigned loff = ldsbase + c * 16;
      unsigned long long ga = (unsigned long long)(xrow + c * 4);
      asm volatile("global_load_async_to_lds_b128 %0, %1, off"
                   :: "v"(loff), "v"(ga) : "memory");
    }
  }

  // --- DFT16 per-lane WMMA fragments (symmetric: A-role for stages 1-2,
  // --- B-role for stage 3). dre/dimm hold {K=kb, K=kb+1} per the 16x4 layout.
  v2f dre[4], dimm[4];
#pragma unroll
  for (int kk = 0; kk < 4; ++kk) {
    int kb = 4 * kk + 2 * half;
    float a0 = -PI2F * (float)((mi * kb) & 15) / 16.0f;
    float a1 = -PI2F * (float)((mi * (kb + 1)) & 15) / 16.0f;
    float s0, c0, s1, c1;
    __sincosf(a0, &s0, &c0);
    __sincosf(a1, &s1, &c1);
    dre[kk].x = c0; dre[kk].y = c1;
    dimm[kk].x = s0; dimm[kk].y = s1;
  }

  asm volatile("s_wait_asynccnt 0x0" ::: "memory");
  __syncthreads();

  // --- Step 0: v = x * w1 (complex, elementwise), raw x from LDS -> S
  const float2* xraw = (const float2*)(smem + 8192);
  const float2* w12  = (const float2*)w1;
#pragma unroll
  for (int i = 0; i < 16; ++i) {
    int n = i * 256 + tid;               // n = n2*256 + n1*16 + n0
    float2 xv = xraw[n];
    float2 wv = w12[n];
    s_re[n] = xv.x * wv.x - xv.y * wv.y;
    s_im[n] = xv.x * wv.y + xv.y * wv.x;
  }
  __syncthreads();

  // --- Stage 1: T[k2*256 + c] = sum_n2 D16[k2][n2] * S[n2*256 + c]
#pragma unroll
  for (int tt = 0; tt < 2; ++tt) {
    int ct = wid * 2 + tt;               // column tile
    int cb = ct * 16 + mi;               // this lane's column (N index)
    v8f cr = {0.f,0.f,0.f,0.f,0.f,0.f,0.f,0.f};
    v8f ci = {0.f,0.f,0.f,0.f,0.f,0.f,0.f,0.f};
#pragma unroll
    for (int kk = 0; kk < 4; ++kk) {
      int k0 = 4 * kk + 2 * half;        // B rows {k0, k0+1} for this lane
      v2f br, bi;
      br.x = s_re[k0 * 256 + cb];       br.y = s_re[(k0 + 1) * 256 + cb];
      bi.x = s_im[k0 * 256 + cb];       bi.y = s_im[(k0 + 1) * 256 + cb];
      cmac4(cr, ci, dre[kk], dimm[kk], br, bi);
    }
#pragma unroll
    for (int v = 0; v < 8; ++v) {        // D: M = v + 8*half, N = mi
      int m = v + 8 * half;
      t_re[m * 256 + cb] = cr[v];
      t_im[m * 256 + cb] = ci[v];
    }
  }
  __syncthreads();

  // --- Twiddle 1: T[k2][n1][n0] *= W_256^{n1*k2}; k2 == i, so use a running
  // --- rotation (W_256^{n1})^i per thread instead of per-element sincos.
  {
    int n1 = tid >> 4;
    float str, sti;
    __sincosf(-PI2F * (float)n1 / 256.0f, &sti, &str);
    float pr = 1.f, pi = 0.f;
    for (int i = 0; i < 16; ++i) {
      int j = i * 256 + tid;
      float re = t_re[j], im = t_im[j];
      t_re[j] = re * pr - im * pi;
      t_im[j] = re * pi + im * pr;
      float nr = pr * str - pi * sti;
      pi = pr * sti + pi * str;
      pr = nr;
    }
  }
  __syncthreads();

  // --- Stage 2: per k2-slab, S[k2*256 + k1*16 + n0] = sum_n1 D16[k1][n1]*T[..]
#pragma unroll
  for (int tt = 0; tt < 2; ++tt) {
    int slab = wid * 2 + tt;             // k2
    v8f cr = {0.f,0.f,0.f,0.f,0.f,0.f,0.f,0.f};
    v8f ci = {0.f,0.f,0.f,0.f,0.f,0.f,0.f,0.f};
#pragma unroll
    for (int kk = 0; kk < 4; ++kk) {
      int k0 = 4 * kk + 2 * half;        // n1 rows {k0, k0+1}
      v2f br, bi;
      br.x = t_re[slab * 256 + k0 * 16 + mi];
      br.y = t_re[slab * 256 + (k0 + 1) * 16 + mi];
      bi.x = t_im[slab * 256 + k0 * 16 + mi];
      bi.y = t_im[slab * 256 + (k0 + 1) * 16 + mi];
      cmac4(cr, ci, dre[kk], dimm[kk], br, bi);
    }
#pragma unroll
    for (int v = 0; v < 8; ++v) {        // D: M = k1, N = n0
      int k1 = v + 8 * half;
      s_re[slab * 256 + k1 * 16 + mi] = cr[v];
      s_im[slab * 256 + k1 * 16 + mi] = ci[v];
    }
  }
  __syncthreads();

  // --- Twiddle 2: S[k2][k1][n0] *= W_4096^{n0*(16*k1+k2)}
  // --- = W_256^{n0*k1} (per-thread constant) * (W_4096^{n0})^i running.
  {
    int n0 = tid & 15, k1 = tid >> 4;
    float a0r, a0i, str, sti;
    __sincosf(-PI2F * (float)((n0 * k1) & 255) / 256.0f, &a0i, &a0r);
    __sincosf(-PI2F * (float)n0 / 4096.0f, &sti, &str);
    float pr = 1.f, pi = 0.f;
    for (int i = 0; i < 16; ++i) {
      int j = i * 256 + tid;
      float fr = a0r * pr - a0i * pi;
      float fi = a0r * pi + a0i * pr;
      float re = s_re[j], im = s_im[j];
      s_re[j] = re * fr - im * fi;
      s_im[j] = re * fi + im * fr;
      float nr = pr * str - pi * sti;
      pi = pr * sti + pi * str;
      pr = nr;
    }
  }
  __syncthreads();

  // --- Stage 3: Y[(k2,k1)][k0] = sum_n0 S[(k2,k1)*16 + n0] * D16[n0][k0].
  // Only k0 < 4 needed (k = 256*k0 + 16*k1 + k2 < 1024). Data is the A
  // operand here; D16 (symmetric) is B. Store Y into T[0..1023].
#pragma unroll
  for (int tt = 0; tt < 2; ++tt) {
    int mt = wid * 2 + tt;               // row tile over (k2,k1)
    v8f cr = {0.f,0.f,0.f,0.f,0.f,0.f,0.f,0.f};
    v8f ci = {0.f,0.f,0.f,0.f,0.f,0.f,0.f,0.f};
#pragma unroll
    for (int kk = 0; kk < 4; ++kk) {
      int ks = 4 * kk + 2 * half;        // A cols {ks, ks+1} for this lane
      v2f ar, ai;
      ar.x = s_re[mt * 256 + mi * 16 + ks];
      ar.y = s_re[mt * 256 + mi * 16 + ks + 1];
      ai.x = s_im[mt * 256 + mi * 16 + ks];
      ai.y = s_im[mt * 256 + mi * 16 + ks + 1];
      cmac4(cr, ci, ar, ai, dre[kk], dimm[kk]);
    }
    if (mi < 4) {                        // N = k0 = mi; discard k0 >= 4
#pragma unroll
      for (int v = 0; v < 8; ++v) {
        int r = mt * 16 + v + 8 * half;  // r = k2*16 + k1
        int k1 = r & 15, k2 = r >> 4;
        int k = mi * 256 + k1 * 16 + k2; // final FFT index, k < 1024
        t_re[k] = cr[v];
        t_im[k] = ci[v];
      }
    }
  }
  __syncthreads();

  // --- Layer 2: h = relu(Y) (re/im separately), h *= w2, then 5-point
  // truncated 1024-DFT: o[j] = sum_k h[k] * W_1024^{k*j}, j = 0..4.
  // W_1024^{(i*256+tid)*j} = W_1024^{tid*j} * W_4^{i*j}; the W_4 powers are
  // exact quarter-turn rotations, applied at compile time (loops unrolled).
  float accr[5] = {0.f, 0.f, 0.f, 0.f, 0.f};
  float acci[5] = {0.f, 0.f, 0.f, 0.f, 0.f};
  float bwr[5], bwi[5];
  bwr[0] = 1.f; bwi[0] = 0.f;
#pragma unroll
  for (int j = 1; j < 5; ++j) {
    float ang = -PI2F * (float)((tid * j) & 1023) / 1024.0f;
    __sincosf(ang, &bwi[j], &bwr[j]);
  }
  const float2* w22 = (const float2*)w2;
#pragma unroll
  for (int i = 0; i < 4; ++i) {
    int k = i * 256 + tid;
    float yr = fmaxf(t_re[k], 0.f);
    float yi = fmaxf(t_im[k], 0.f);
    float2 wv = w22[k];
    float hr = yr * wv.x - yi * wv.y;
    float hi = yr * wv.y + yi * wv.x;
#pragma unroll
    for (int j = 0; j < 5; ++j) {
      float tr = hr * bwr[j] - hi * bwi[j];
      float ti = hr * bwi[j] + hi * bwr[j];
      const int q = (i * j) & 3;         // multiply by W_4^q = (-i)^q, exact
      float rr = (q == 0) ? tr : (q == 1) ? ti : (q == 2) ? -tr : -ti;
      float ri = (q == 0) ? ti : (q == 1) ? -tr : (q == 2) ? -ti : tr;
      accr[j] += rr;
      acci[j] += ri;
    }
  }
  // wave32 tree reduction
#pragma unroll
  for (int off = 16; off > 0; off >>= 1) {
#pragma unroll
    for (int j = 0; j < 5; ++j) {
      accr[j] += __shfl_down(accr[j], off, 32);
      acci[j] += __shfl_down(acci[j], off, 32);
    }
  }
  float* red = s_re;                     // S is dead; reuse as scratch
  if (lane == 0) {
#pragma unroll
    for (int j = 0; j < 5; ++j) {
      red[wid * 10 + 2 * j]     = accr[j];
      red[wid * 10 + 2 * j + 1] = acci[j];
    }
  }
  __syncthreads();
  if (tid < 10) {
    float s = 0.f;
#pragma unroll
    for (int w = 0; w < 8; ++w) s += red[w * 10 + tid];
    out[row * 10 + tid] = s;             // [B, 10, 1] flat, interleaved re/im
  }
}

extern "C" void kernel_launch(void* const* d_in, const int* in_sizes, int n_in,
                              void* d_out, int out_size, void* d_ws,
                              size_t ws_size, hipStream_t stream) {
  (void)in_sizes; (void)n_in; (void)d_ws; (void)ws_size; (void)out_size;
  const float* x  = (const float*)d_in[0];   // [8192, 8192]
  const float* w1 = (const float*)d_in[1];   // [8192] (4096 complex)
  const float* w2 = (const float*)d_in[2];   // [2048] (1024 complex)
  float* out = (float*)d_out;                // [8192 * 10]
  fftmlp_kernel<<<dim3(8192), dim3(256), 0, stream>>>(x, w1, w2, out);
}